// BinaryMapper_74122545594587
// MI455X (gfx1250) — compile-verified
//
#include <hip/hip_runtime.h>

typedef __attribute__((ext_vector_type(2))) float v2f;
typedef __attribute__((ext_vector_type(4))) float v4f;
typedef __attribute__((ext_vector_type(8))) float v8f;

#define NUM_BITS   14
#define NUM_CAT    16384
#define HALF_BITS  7
#define HALF_CAT   128    // 2^7

__device__ __forceinline__ float sigmoid_f(float x) {
    return 1.0f / (1.0f + __expf(-x));
}

__device__ __forceinline__ unsigned hash_u32(unsigned x) {
    x ^= x >> 16; x *= 0x7feb352dU;
    x ^= x >> 15; x *= 0x846ca68bU;
    x ^= x >> 16;
    return x;
}

// One block (8 wave32s) per token.
//   probs (viewed 128x128: row = chi hi-7-bits, col = clo lo-7-bits) = outer(Phi, Plo)
//   computed as an 8x8 grid of 16x16 tiles via rank-1 V_WMMA_F32_16X16X4_F32.
__global__ __launch_bounds__(256) void binary_mapper_kernel(
    const float* __restrict__ logits,    // (n_tokens, 14)
    float*       __restrict__ out_idx,   // (n_tokens)   sampled index as float
    float*       __restrict__ out_probs, // (n_tokens, 16384)
    float*       __restrict__ out_onehot)// (n_tokens, 16384)  == st forward value
{
    __shared__ float sX[NUM_BITS];
    __shared__ float sPhi[HALF_CAT];
    __shared__ float sPlo[HALF_CAT];
    __shared__ int   sIndex;

    const int token = blockIdx.x;
    const int t     = threadIdx.x;

    if (t < NUM_BITS) sX[t] = logits[token * NUM_BITS + t];
    __syncthreads();

    // Per-half products of Bernoulli factors (MSB-first bit order matches
    // the reference codebook: codebook[c,h] = bit (13-h) of c).
    if (t < HALF_CAT) {
        const int chi = t;
        float v = 1.0f;
#pragma unroll
        for (int h = 0; h < HALF_BITS; ++h) {
            const float p = sigmoid_f(sX[h]);
            v *= ((chi >> (HALF_BITS - 1 - h)) & 1) ? p : (1.0f - p);
        }
        sPhi[chi] = v;
    } else {
        const int clo = t - HALF_CAT;
        float v = 1.0f;
#pragma unroll
        for (int h = 0; h < HALF_BITS; ++h) {
            const float p = sigmoid_f(sX[HALF_BITS + h]);
            v *= ((clo >> (HALF_BITS - 1 - h)) & 1) ? p : (1.0f - p);
        }
        sPlo[clo] = v;
    }

    if (t == 0) {
        // Factorized categorical sample: 14 independent Bernoulli draws,
        // deterministic per (token, bit).
        unsigned idx = 0;
#pragma unroll
        for (int h = 0; h < NUM_BITS; ++h) {
            const float p = sigmoid_f(sX[h]);
            const unsigned r = hash_u32(0x9e3779b9u ^ (unsigned)(token * 0x85ebca6b)
                                                   ^ (unsigned)(h * 0xc2b2ae35) ^ 42u);
            const float u = (float)(r >> 8) * (1.0f / 16777216.0f);
            idx = (idx << 1) | (u < p ? 1u : 0u);
        }
        sIndex = (int)idx;
        out_idx[token] = (float)idx;
    }
    __syncthreads();

    // ---- rank-1 outer product via WMMA f32 16x16x4 ----
    const int  wave    = t >> 5;       // 0..7 : tile-row (chi block)
    const int  lane    = t & 31;
    const int  laneN   = lane & 15;
    const bool lowHalf = lane < 16;
    const float zero   = 0.0f;

    // A (16x4 f32): lanes 0-15 VGPR0 hold K=0 -> Phi chunk; all other K zero.
    // Load unconditionally (index valid for every lane), then select: avoids
    // exec-masked ds_load branches, leaves a single v_cndmask.
    const float phi = sPhi[wave * 16 + laneN];
    v2f a;
    a.x = lowHalf ? phi : zero;
    a.y = zero;

    float* probs_tok = out_probs + (size_t)token * NUM_CAT;
    // D layout: VGPR r -> row M = r (lanes 0-15) or r+8 (lanes 16-31), N = lane%16.
    const int rowBase = wave * 16 + (lowHalf ? 0 : 8);

#pragma unroll
    for (int nlo = 0; nlo < 8; ++nlo) {
        // B (4x16 f32): K=0 row (lanes 0-15, VGPR0) = Plo chunk; rest zero.
        const float plo = sPlo[nlo * 16 + laneN];   // unconditional LDS read
        v2f b;
        b.x = lowHalf ? plo : zero;
        b.y = zero;
        v8f c = {0.f, 0.f, 0.f, 0.f, 0.f, 0.f, 0.f, 0.f};
        // D[m,n] = A[m,0]*B[0,n] = Phi[m] * Plo[n]
        v8f d = __builtin_amdgcn_wmma_f32_16x16x4_f32(
            /*neg_a=*/false, a, /*neg_b=*/false, b,
            /*c_mod=*/(short)0, c, /*reuse_a=*/false, /*reuse_b=*/false);
#pragma unroll
        for (int r = 0; r < 8; ++r) {
            // Streaming output (never re-read; 268 MB total > 192 MB L2):
            // non-temporal store keeps L2 from being thrashed.
            __builtin_nontemporal_store(
                d[r], probs_tok + (size_t)(rowBase + r) * HALF_CAT + nlo * 16 + laneN);
        }
    }

    // ---- st forward value == one_hot(sampled) : pure b128 NT stores ----
    float* oh_tok = out_onehot + (size_t)token * NUM_CAT;
    const int sampled = sIndex;
#pragma unroll
    for (int i = 0; i < NUM_CAT / (256 * 4); ++i) {   // 16 iterations of float4
        const int c0 = i * 1024 + t * 4;
        v4f v = {0.0f, 0.0f, 0.0f, 0.0f};
        const unsigned off = (unsigned)(sampled - c0);
        if (off < 4u) v[off] = 1.0f;
        __builtin_nontemporal_store(v, (v4f*)(oh_tok + c0));
    }
}

extern "C" void kernel_launch(void* const* d_in, const int* in_sizes, int n_in,
                              void* d_out, int out_size, void* d_ws, size_t ws_size,
                              hipStream_t stream) {
    const float* logits = (const float*)d_in[0];
    const int n_tokens  = in_sizes[0] / NUM_BITS;   // 2*1024 = 2048

    float* out        = (float*)d_out;
    float* out_idx    = out;                                   // n_tokens
    float* out_probs  = out + n_tokens;                        // n_tokens*16384
    float* out_onehot = out_probs + (size_t)n_tokens * NUM_CAT;// n_tokens*16384

    binary_mapper_kernel<<<n_tokens, 256, 0, stream>>>(logits, out_idx, out_probs, out_onehot);
}